// ctc_loss_43379169689615
// MI455X (gfx1250) — compile-verified
//
#include <hip/hip_runtime.h>
#include <hip/hip_bf16.h>
#include <stdint.h>

// Problem constants (reference): B=64, T=1024, V=512, L=128, BLANK=0
#define B_ 64
#define T_ 1024
#define V_ 512
#define L_ 128
#define S_ (2 * L_ + 1)   // 257 extended states
#define BLOCK_ 288        // 9 wave32s (states 0..256 + 31 idle lanes)
#define NW_ 9             // waves per block
#define K_ 32             // emission prefetch depth (MUST match s_wait_asynccnt imm below)
#define NEGINF (-1e30f)

static_assert(K_ == 32, "s_wait_asynccnt immediate is hardcoded to K_-1 = 31");

typedef __attribute__((ext_vector_type(2))) float v2f;
typedef __attribute__((ext_vector_type(8))) float v8f;

// CDNA5 async gather: per-lane global load written directly into LDS, ASYNCcnt-tracked.
__device__ __forceinline__ void async_gather_b32(unsigned lds_addr, const float* gptr) {
  asm volatile("global_load_async_to_lds_b32 %0, %1, off"
               :: "v"(lds_addr), "v"(gptr) : "memory");
}
// Pop the async pipeline to <= 31 outstanding: with exactly 32 in flight and
// in-order completion, this guarantees the oldest gather has landed in LDS.
__device__ __forceinline__ void wait_async_le31() {
  asm volatile("s_wait_asynccnt 31" ::: "memory");
}

__global__ __launch_bounds__(BLOCK_) void ctc_alpha_pipelined(
    const float* __restrict__ outputs,      // (B,T,V) treated as log-probs by the reference
    const int*   __restrict__ labels,       // (B,L)
    const int*   __restrict__ out_len,      // (B,)
    const int*   __restrict__ lab_len,      // (B,)
    float*       __restrict__ losses)       // (B,)
{
  const int b    = blockIdx.x;
  const int s    = threadIdx.x;
  const int w    = s >> 5;                  // wave id 0..8
  const int lane = s & 31;
  const bool active = (s < S_);

  // Full-depth boundary ring: producer wave w publishes its top-2 alphas for
  // every t; depth T_ => producers can run arbitrarily ahead, no back-pressure.
  __shared__ float ring[NW_ - 1][T_][2];    // 64 KB
  __shared__ int   progress[NW_ - 1];       // last published t per producer wave
  __shared__ float emitb[K_][BLOCK_];       // 36 KB async-gathered emissions
  __shared__ float fin[2];

  if (s < NW_ - 1) progress[s] = -1;

  // Extended label (even -> BLANK, odd -> labels[s/2]) and the skip rule.
  int lab = 0; bool skip = false;
  if (active && (s & 1)) {
    lab = labels[b * L_ + (s >> 1)];
    if (s >= 3) {
      int lp = labels[b * L_ + (s >> 1) - 1];
      skip = (lab != 0) && (lab != lp);
    }
  }
  const int Tb = out_len[b];
  const int Lb = lab_len[b];

  const float*   gp0     = outputs + (size_t)b * T_ * V_ + lab;  // this state's column
  const unsigned lbase   = (unsigned)(uintptr_t)(&emitb[0][s]);  // DS addr = low 32 bits
  const unsigned lstride = (unsigned)(sizeof(float) * BLOCK_);

  // t = 0: alpha0[s] = emit(0,s) for s<2 else NEG (direct load, not pipelined).
  float a = active ? gp0[0] : NEGINF;
  if (s >= 2) a = NEGINF;

  // Prologue: put exactly K_ gathers in flight (rows clamped to Tb-1; the
  // clamped duplicates land in slots that are never read -> harmless).
  if (active) {
#pragma unroll
    for (int tp = 1; tp <= K_; ++tp) {
      int tt = (tp < Tb) ? tp : (Tb - 1);
      async_gather_b32(lbase + (unsigned)(tp & (K_ - 1)) * lstride,
                       gp0 + (size_t)tt * V_);
    }
  }
  __syncthreads();                          // progress[] visible to all waves

  // Publish t = 0 boundaries.
  volatile int* prog = progress;
  if (w < NW_ - 1) {
    if (lane >= 30) ring[w][0][lane - 30] = a;
    asm volatile("" ::: "memory");          // data before flag (DS is in-order per wave)
    if (lane == 31) prog[w] = 0;
  }

  for (int t = 1; t < Tb; ++t) {
    // ---- emission for step t (issued K_ steps ago) ----
    wait_async_le31();
    float e = active ? emitb[t & (K_ - 1)][s] : 0.0f;
    // Refill: gather for step t+K_ into the slot just consumed (clamped tail).
    if (active) {
      int tt = (t + K_ < Tb) ? (t + K_) : (Tb - 1);
      async_gather_b32(lbase + (unsigned)(t & (K_ - 1)) * lstride,
                       gp0 + (size_t)tt * V_);
    }

    // ---- boundary alphas (states w*32-2, w*32-1 at t-1) from wave w-1 ----
    float b0 = NEGINF, b1 = NEGINF;
    if (w > 0) {
      while (prog[w - 1] < t - 1) { /* spin; producer never waits on us */ }
      asm volatile("" ::: "memory");        // no hoisting of ring reads above spin
      b0 = ring[w - 1][t - 1][0];
      b1 = ring[w - 1][t - 1][1];
    }

    // ---- 3-way LSE recursion, alpha kept in registers, neighbors via wave32 shuffles ----
    float p0 = a;
    float p1 = __shfl_up(a, 1, 32);
    float p2 = __shfl_up(a, 2, 32);
    if (lane == 0) { p1 = b1; p2 = b0; }
    if (lane == 1) { p2 = b1; }
    if (!skip) p2 = NEGINF;
    float m   = fmaxf(p0, fmaxf(p1, p2));
    float lse = m + __logf(__expf(p0 - m) + __expf(p1 - m) + __expf(p2 - m));
    a = lse + e;                            // alpha at time t for this state

    // ---- publish boundaries for wave w+1 ----
    if (w < NW_ - 1) {
      if (lane >= 30) ring[w][t][lane - 30] = a;
      asm volatile("" ::: "memory");
      if (lane == 31) prog[w] = t;
    }
  }

  // Final 2-state LSE at t = Tb-1: states 2*Lb-1 and 2*Lb.
  if (active && s == 2 * Lb - 1) fin[0] = a;
  if (active && s == 2 * Lb)     fin[1] = a;
  __syncthreads();
  if (s == 0) {
    float a0 = fin[0], a1 = fin[1];
    float m  = fmaxf(a0, a1);
    losses[b] = -(m + __logf(__expf(a0 - m) + __expf(a1 - m))) / (float)Lb;
  }
}

// Batch-mean via V_WMMA_F32_16X16X4_F32 (verified lowering in round 1).
// With B(4x16)=all-ones, every D entry is a row-sum of A, so
// sum(all D) = 16 * sum(all A) for ANY bijective packing of the 64 losses.
__global__ __launch_bounds__(32) void ctc_reduce_wmma(
    const float* __restrict__ losses, float* __restrict__ out)
{
  const int lane = threadIdx.x;
  v2f av; av.x = losses[2 * lane]; av.y = losses[2 * lane + 1];
  v2f bv; bv.x = 1.0f; bv.y = 1.0f;
  v8f c = {0.f, 0.f, 0.f, 0.f, 0.f, 0.f, 0.f, 0.f};
  c = __builtin_amdgcn_wmma_f32_16x16x4_f32(false, av, false, bv, (short)0, c, false, false);
  float p = c[0] + c[1] + c[2] + c[3] + c[4] + c[5] + c[6] + c[7];
#pragma unroll
  for (int off = 16; off > 0; off >>= 1) p += __shfl_xor(p, off, 32);
  if (lane == 0) out[0] = p * (1.0f / (16.0f * (float)B_));
}

extern "C" void kernel_launch(void* const* d_in, const int* in_sizes, int n_in,
                              void* d_out, int out_size, void* d_ws, size_t ws_size,
                              hipStream_t stream) {
  (void)in_sizes; (void)n_in; (void)out_size; (void)ws_size;
  const float* outputs = (const float*)d_in[0];
  const int*   labels  = (const int*)d_in[1];
  const int*   olen    = (const int*)d_in[2];
  const int*   llen    = (const int*)d_in[3];
  float* losses = (float*)d_ws;             // 64 floats of scratch
  float* out    = (float*)d_out;

  ctc_alpha_pipelined<<<dim3(B_), dim3(BLOCK_), 0, stream>>>(outputs, labels, olen, llen, losses);
  ctc_reduce_wmma<<<dim3(1), dim3(32), 0, stream>>>(losses, out);
}